// AffinityHead_28398323761860
// MI455X (gfx1250) — compile-verified
//
#include <hip/hip_runtime.h>

typedef __attribute__((ext_vector_type(2))) float v2f;
typedef __attribute__((ext_vector_type(8))) float v8f;

#define CCH   128   // channels
#define KDIM  258   // 2*C + 2
#define NG    65    // K groups of 4 (64 for src+dst, 1 for delta+zero pad)
#define NT    8     // number of 16-wide N tiles (128 / 16)

// LDS layout: w1p is [NG][256] float2 (g-major), where
//   w1p[g][n*2 + h] = { w1[4g+2h][n], w1[4g+2h+1][n] }   (rows >= 258 stored as 0)
// A lane (half h, lm) loads B frag for (g, tile t) at float2 index  g*256 + 32t + (2*lm + h):
//   - t stride = 32 float2 = 256 B   -> fits DS immediate offsets
//   - g stride = 256 float2 = 2048 B -> one running-pointer bump per K-step
//   - bank = 4*lm + 2*h (mod 64)     -> conflict-free b64 loads across the wave

__global__ __launch_bounds__(256) void affinity_wmma_kernel(
    const float* __restrict__ tokens,   // [N,128]
    const float* __restrict__ coords,   // [N,2]
    const int*   __restrict__ ei,       // [2,E]
    const float* __restrict__ w1,       // [258,128]
    const float* __restrict__ b1,       // [128]
    const float* __restrict__ w2,       // [128]
    const float* __restrict__ b2,       // [1]
    float* __restrict__ out,            // [E]
    int E)
{
    extern __shared__ float smem[];
    float* w1p = smem;                     // NG*512 floats (NG*256 float2)
    float* b1s = smem + NG * 512;          // [128]
    float* w2s = b1s + CCH;                // [128]

    // ---- stage packed weights into LDS ----
    for (int i = threadIdx.x; i < NG * 256; i += blockDim.x) {
        int g = i >> 8;            // K group
        int h = (i >> 7) & 1;      // lane half this pair serves
        int n = i & 127;           // output channel
        int k0 = 4 * g + 2 * h;
        v2f v;
        v.x = (k0     < KDIM) ? w1[k0 * CCH + n]       : 0.0f;
        v.y = (k0 + 1 < KDIM) ? w1[(k0 + 1) * CCH + n] : 0.0f;
        *(v2f*)&w1p[(g * 256 + n * 2 + h) * 2] = v;
    }
    for (int i = threadIdx.x; i < CCH; i += blockDim.x) {
        b1s[i] = b1[i];
        w2s[i] = w2[i];
    }
    __syncthreads();

    const int lane  = threadIdx.x & 31;
    const int half  = lane >> 4;        // 0: lanes 0-15, 1: lanes 16-31
    const int lm    = lane & 15;        // row-in-tile (A/M) and col-in-Ntile (B/N)
    const int wave  = (blockIdx.x * blockDim.x + threadIdx.x) >> 5;
    const int nWaves = (gridDim.x * blockDim.x) >> 5;
    const int nTiles = (E + 15) >> 4;
    const float b2v = b2[0];
    const int kb = half * 2;            // lane-half K offset inside each K=4 group

    // lane bases into packed B (float index); frag(g,t) at  base + g*512 + t*64.
    // The 32*512 / 64*512 float biases are laundered as opaque *integers* so the
    // compiler cannot re-fold them into per-load DS offsets (they are one past the
    // 16-bit immediate range), while the pointer chain stays rooted at smem so
    // address-space inference still lowers these to native ds_load (not flat).
    const float* bbaseS = w1p + (2 * lm + half) * 2;      // K groups  0..31
    int baseD = 32 * 512;                                  // float offset of dst groups
    int baseT = 64 * 512;                                  // float offset of delta group
    asm("" : "+v"(baseD));
    asm("" : "+v"(baseT));
    const float* bbaseD = bbaseS + baseD;                  // K groups 32..63
    const float* bbaseT = bbaseS + baseT;                  // K group  64 (delta)

    // tile-invariant epilogue fragments
    float b1f[NT], w2f[NT];
    #pragma unroll
    for (int t = 0; t < NT; ++t) {
        b1f[t] = b1s[t * 16 + lm];
        w2f[t] = w2s[t * 16 + lm];
    }

    for (int tile = wave; tile < nTiles; tile += nWaves) {
        int e  = tile * 16 + lm;
        int ec = e < E ? e : (E - 1);
        int s  = ei[ec];
        int d  = ei[E + ec];
        const float* rowS = tokens + (size_t)s * CCH + kb;
        const float* rowD = tokens + (size_t)d * CCH + kb;

        v8f acc[NT];
        #pragma unroll
        for (int t = 0; t < NT; ++t) acc[t] = (v8f)0.0f;

        // ---- src half: K groups 0..31 (K = 0..127) ----
        #pragma unroll 2
        for (int g = 0; g < 32; ++g) {
            v2f a = *(const v2f*)(rowS + g * 4);        // A frag: {K, K+1} of row M=lm
            #pragma unroll
            for (int t = 0; t < NT; ++t) {
                v2f b = *(const v2f*)(bbaseS + g * 512 + t * 64);
                acc[t] = __builtin_amdgcn_wmma_f32_16x16x4_f32(
                             false, a, false, b, (short)0, acc[t], false, false);
            }
        }

        // ---- dst half: K groups 32..63 (K = 128..255) ----
        #pragma unroll 2
        for (int g = 0; g < 32; ++g) {
            v2f a = *(const v2f*)(rowD + g * 4);
            #pragma unroll
            for (int t = 0; t < NT; ++t) {
                v2f b = *(const v2f*)(bbaseD + g * 512 + t * 64);
                acc[t] = __builtin_amdgcn_wmma_f32_16x16x4_f32(
                             false, a, false, b, (short)0, acc[t], false, false);
            }
        }

        // ---- Δcoord rank-2 update: K group 64 (K=256,257 real; 258,259 are zero rows) ----
        // B rows 258/259 are stored as exact zeros, so A needs no lane masking:
        // lanes 16-31 contribute delta * 0 = 0.
        {
            v2f cs = *(const v2f*)(coords + 2 * s);
            v2f cd = *(const v2f*)(coords + 2 * d);
            v2f a;
            a.x = cd.x - cs.x;
            a.y = cd.y - cs.y;
            #pragma unroll
            for (int t = 0; t < NT; ++t) {
                v2f b = *(const v2f*)(bbaseT + t * 64);
                acc[t] = __builtin_amdgcn_wmma_f32_16x16x4_f32(
                             false, a, false, b, (short)0, acc[t], false, false);
            }
        }

        // ---- epilogue: bias + ReLU + dot(w2), reduce over N within each 16-lane half ----
        // C layout: acc[t][r] holds element (M = r + 8*half, N = 16*t + lm)
        float partial[8];
        #pragma unroll
        for (int r = 0; r < 8; ++r) partial[r] = 0.0f;
        #pragma unroll
        for (int t = 0; t < NT; ++t) {
            #pragma unroll
            for (int r = 0; r < 8; ++r) {
                float h = acc[t][r] + b1f[t];
                h = h > 0.0f ? h : 0.0f;
                partial[r] += h * w2f[t];
            }
        }
        #pragma unroll
        for (int r = 0; r < 8; ++r) {
            partial[r] += __shfl_xor(partial[r], 8, 32);
            partial[r] += __shfl_xor(partial[r], 4, 32);
            partial[r] += __shfl_xor(partial[r], 2, 32);
            partial[r] += __shfl_xor(partial[r], 1, 32);
        }
        if (lm == 0) {
            #pragma unroll
            for (int r = 0; r < 8; ++r) {
                int eo = tile * 16 + half * 8 + r;      // lanes 0 / 16 write rows 0-7 / 8-15
                if (eo < E) out[eo] = partial[r] + b2v;
            }
        }
    }
}

extern "C" void kernel_launch(void* const* d_in, const int* in_sizes, int n_in,
                              void* d_out, int out_size, void* d_ws, size_t ws_size,
                              hipStream_t stream) {
    const float* tokens = (const float*)d_in[0];
    const float* coords = (const float*)d_in[1];
    const int*   ei     = (const int*)d_in[2];
    const float* w1     = (const float*)d_in[3];
    const float* b1     = (const float*)d_in[4];
    const float* w2     = (const float*)d_in[5];
    const float* b2     = (const float*)d_in[6];
    float* out = (float*)d_out;

    const int E = in_sizes[2] / 2;
    if (E <= 0) return;

    const size_t lds_bytes = (size_t)(NG * 512 + 2 * CCH) * sizeof(float);  // ~131.4 KB
    hipFuncSetAttribute(reinterpret_cast<const void*>(affinity_wmma_kernel),
                        hipFuncAttributeMaxDynamicSharedMemorySize, (int)lds_bytes);

    dim3 grid(512), block(256);
    hipLaunchKernelGGL(affinity_wmma_kernel, grid, block, lds_bytes, stream,
                       tokens, coords, ei, w1, b1, w2, b2, out, E);
}